// Module_1_18605798326828
// MI455X (gfx1250) — compile-verified
//
#include <hip/hip_runtime.h>

typedef __attribute__((ext_vector_type(2))) float v2f;
typedef __attribute__((ext_vector_type(8))) float v8f;

struct Params {
    const float* W[10];
    const float* B[10];
};

// ---- CDNA5 WMMA: D(16x16 f32) = A(16x4 f32) * B(4x16 f32) + C ----
__device__ __forceinline__ v8f wmma4(v2f a, v2f b, v8f c) {
    return __builtin_amdgcn_wmma_f32_16x16x4_f32(
        /*neg_a=*/false, a, /*neg_b=*/false, b,
        /*c_mod=*/(short)0, c, /*reuse_a=*/false, /*reuse_b=*/false);
}

// lane <-> lane^16 full-wave exchange via v_permlanex16_b32 (pure VALU,
// identity nibble selects -> lane i^16).
__device__ __forceinline__ float swap16(float x) {
    int xi = __float_as_int(x);
    int r  = __builtin_amdgcn_permlanex16(xi, xi, 0x76543210, 0xFEDCBA98,
                                          /*fi=*/false, /*bound_ctrl=*/false);
    return __int_as_float(r);
}

// Single-instruction ReLU.  fmed3/fmaxf get re-folded by LLVM into a
// canonicalize + max pair; raw asm pins it to exactly one v_max_num_f32.
__device__ __forceinline__ float relu_asm(float x) {
    float y;
    asm("v_max_num_f32 %0, 0, %1" : "=v"(y) : "v"(x));
    return y;
}

// Value of h[k] (K real features, k==K -> bias-one, k>K -> 0) for this lane,
// given previous layer's raw D accumulator (lane = sample, VGPR r = feature m,
// lanes 16..31 hold m = r+8).  HI = this lane is in the upper half.
template<int KK, int K, bool HI>
__device__ __forceinline__ float valfor(const v8f& d) {
    if constexpr (KK > K) {
        return 0.0f;
    } else if constexpr (KK == K) {
        return 1.0f;               // constant-1 feature carrying the bias
    } else {
        constexpr int r = KK & 7;
        float v = d[r];
        constexpr bool direct = HI ? (KK >= 8) : (KK < 8);
        if constexpr (direct) return v;
        else                  return swap16(v);
    }
}

// One B-matrix VGPR: low-half lanes carry k=KL, high-half lanes carry k=KL+2.
// ReLU is applied at consumption (each hidden activation is read exactly
// once).  relu(1)=1 and relu(0)=0, so relu commutes with the half-select;
// apply the asm max only to the non-constant side(s) to keep constant
// folding on bias/pad slots.
template<int KL, int K, bool RELU>
__device__ __forceinline__ float bslice(const v8f& d, bool hi) {
    constexpr bool loReal = (KL     < K);
    constexpr bool hiReal = (KL + 2 < K);
    float vlo = valfor<KL,     K, false>(d);
    float vhi = valfor<KL + 2, K, true >(d);
    if constexpr (RELU && loReal && hiReal) {
        float v = hi ? vhi : vlo;
        return relu_asm(v);                 // one op covers both halves
    } else if constexpr (RELU && loReal) {
        float v = relu_asm(vlo);            // hi side is a constant
        return hi ? vhi : v;
    } else {
        return hi ? vhi : vlo;              // all-constant (or no relu)
    }
}

static __device__ __forceinline__ v8f zero8() {
    v8f z = {0.f,0.f,0.f,0.f,0.f,0.f,0.f,0.f};
    return z;
}

// One layer: h_next^T = W * [relu(h)^T; 1] using ceil((K+1)/4) chained WMMAs.
// Takes the RAW pre-activation D of the previous layer.
template<int K>
__device__ __forceinline__ v8f gemmK(const v2f* w, const v8f& hprev, bool hi) {
    constexpr int NS = (K + 4) / 4;
    v8f acc = zero8();
    {
        v2f b; b.x = bslice<0, K, true>(hprev, hi); b.y = bslice<1, K, true>(hprev, hi);
        acc = wmma4(w[0], b, acc);
    }
    if constexpr (NS > 1) {
        v2f b; b.x = bslice<4, K, true>(hprev, hi); b.y = bslice<5, K, true>(hprev, hi);
        acc = wmma4(w[1], b, acc);
    }
    if constexpr (NS > 2) {
        v2f b; b.x = bslice<8, K, true>(hprev, hi); b.y = bslice<9, K, true>(hprev, hi);
        acc = wmma4(w[2], b, acc);
    }
    return acc;
}

__global__ __launch_bounds__(256) void mlp_wmma_kernel(
    const float* __restrict__ X, float* __restrict__ Y, int n, Params P) {

    const int lane = threadIdx.x & 31;
    const bool hi  = lane >= 16;
    const int m16  = lane & 15;
    const int wave   = (int)((blockIdx.x * blockDim.x + threadIdx.x) >> 5);
    const int nwaves = (int)((gridDim.x * blockDim.x) >> 5);

    // ---- preload all weights into A-matrix register layout, bias folded ----
    // A 16x4 f32 layout: lane 0-15 -> M, VGPR j holds K = 4s + j + 2*(lane/16)
    v2f w[26];
    {
        const int DIMS[11] = {1,5,10,10,10,10,10,10,10,5,1};
        int idx = 0;
#pragma unroll
        for (int L = 0; L < 10; ++L) {
            const int fin  = DIMS[L];
            const int fout = DIMS[L + 1];
            const int ns   = (fin + 4) / 4;   // covers fin real cols + bias col
#pragma unroll
            for (int s = 0; s < ns; ++s) {
#pragma unroll
                for (int j = 0; j < 2; ++j) {
                    const int k = 4 * s + j + (hi ? 2 : 0);
                    float v = 0.0f;
                    if (m16 < fout) {
                        if (k < fin)       v = P.W[L][m16 * fin + k];
                        else if (k == fin) v = P.B[L][m16];
                    }
                    w[idx][j] = v;
                }
                ++idx;
            }
        }
    }

    // ---- grid-stride over tiles of 32 samples (two 16-sample WMMA chains) --
    for (int base = wave * 32; base < n; base += nwaves * 32) {
        const int gi = base + lane;
        // Branch-free clamped streaming load (n is a multiple of 32 anyway;
        // clamping only dedups the tail, never fabricates out-of-range data).
        const int gl = gi < n ? gi : (n - 1);
        const float xv = __builtin_nontemporal_load(X + gl);
        const float xs = swap16(xv);    // samples base+16.. into lanes 0-15

        v8f h0, h1;   // RAW pre-activation outputs; ReLU applied at consumption
        {   // Layer 1: K=1 (+bias), single WMMA per tile; x is not relu'd
            v2f b0; b0.x = hi ? 0.0f : xv; b0.y = hi ? 0.0f : 1.0f;
            v2f b1; b1.x = hi ? 0.0f : xs; b1.y = hi ? 0.0f : 1.0f;
            h0 = wmma4(w[0], b0, zero8());
            h1 = wmma4(w[0], b1, zero8());
        }
        // Layer 2: 5 -> 10   (relu(h1) consumed inside)
        h0 = gemmK<5>(w + 1, h0, hi);
        h1 = gemmK<5>(w + 1, h1, hi);
        // Layers 3..8: 10 -> 10
#pragma unroll
        for (int t = 0; t < 6; ++t) {
            h0 = gemmK<10>(w + 3 + 3 * t, h0, hi);
            h1 = gemmK<10>(w + 3 + 3 * t, h1, hi);
        }
        // Layer 9: 10 -> 5
        h0 = gemmK<10>(w + 21, h0, hi);
        h1 = gemmK<10>(w + 21, h1, hi);
        // Layer 10: 5 -> 1 (consumes relu(h9); its own output stays linear)
        h0 = gemmK<5>(w + 24, h0, hi);
        h1 = gemmK<5>(w + 24, h1, hi);

        // y[m=0] sits in D VGPR0 lanes 0-15 of each tile.  Move tile1's row
        // into lanes 16-31 and do one fully-coalesced 128B store.
        const float y1 = swap16(h1[0]);
        const float yv = hi ? y1 : h0[0];
        if (base + 32 <= n) {               // wave-uniform fast path
            __builtin_nontemporal_store(yv, Y + gi);
        } else if (gi < n) {                // ragged tail (never hit when 32 | n)
            __builtin_nontemporal_store(yv, Y + gi);
        }
    }
}

extern "C" void kernel_launch(void* const* d_in, const int* in_sizes, int n_in,
                              void* d_out, int out_size, void* d_ws, size_t ws_size,
                              hipStream_t stream) {
    (void)n_in; (void)d_ws; (void)ws_size; (void)out_size;
    const float* X = (const float*)d_in[0];
    float* Y = (float*)d_out;
    const int n = in_sizes[0];

    Params P;
    for (int i = 0; i < 10; ++i) {
        P.W[i] = (const float*)d_in[1 + 2 * i];
        P.B[i] = (const float*)d_in[2 + 2 * i];
    }

    const int threads = 256;                  // 8 wave32 per block
    int tiles32 = (n + 31) / 32;              // 125,000 for n = 4M
    long long waves_wanted = tiles32 < 8192 ? tiles32 : 8192;
    int blocks = (int)((waves_wanted * 32 + threads - 1) / threads);
    if (blocks < 1) blocks = 1;

    mlp_wmma_kernel<<<blocks, threads, 0, stream>>>(X, Y, n, P);
}